// Paillier_Dec_84370337562658
// MI455X (gfx1250) — compile-verified
//
#include <hip/hip_runtime.h>
#include <stdint.h>

// ---------------- Paillier constants (compile-time) ----------------
// N = 3599, N^2 = 12952801 (odd, < 2^24) -> Montgomery with R = 2^32.
#define NPUB  3599u
#define N2PUB 12952801u

// -N2^{-1} mod 2^32 via Newton iteration (x_{k+1} = x_k (2 - n x_k)).
constexpr uint32_t neg_inv_mod_2_32(uint32_t n) {
    uint32_t x = n;                       // correct mod 2^3 for odd n
    for (int i = 0; i < 5; ++i) x *= 2u - n * x;
    return 0u - x;
}
constexpr uint32_t NINV = neg_inv_mod_2_32(N2PUB);
constexpr uint32_t R1   = (uint32_t)((1ull << 32) % N2PUB);           // R   mod N^2
constexpr uint32_t R2   = (uint32_t)(((uint64_t)R1 * R1) % N2PUB);    // R^2 mod N^2

// MU = 870^{-1} mod 3599 (extended Euclid, constexpr)
constexpr uint32_t modinv(uint32_t a, uint32_t n) {
    int64_t t = 0, nt = 1, r = (int64_t)n, nr = (int64_t)a;
    while (nr != 0) {
        int64_t q = r / nr;
        int64_t tmp = t - q * nt; t = nt; nt = tmp;
        tmp = r - q * nr; r = nr; nr = tmp;
    }
    return (uint32_t)(t < 0 ? t + (int64_t)n : t);
}
constexpr uint32_t MUC = modinv(870u, NPUB);

// ---------------- Montgomery modmul: a*b*R^{-1} mod N^2 ----------------
// a,b < N^2 < 2^24. t = a*b < 2^48. u = (t + m*N2)/2^32 < 2*N2.
// lo(t)+lo(m*N2) is 0 or 2^32 exactly, so the carry is (tl != 0).
__device__ __forceinline__ uint32_t montmul(uint32_t a, uint32_t b) {
    uint32_t tl = a * b;
    uint32_t th = __umulhi(a, b);
    uint32_t m  = tl * NINV;
    uint32_t mh = __umulhi(m, N2PUB);
    uint32_t u  = th + mh + (tl != 0u ? 1u : 0u);
    return (u >= N2PUB) ? (u - N2PUB) : u;
}

// Full decrypt of one residue. lambda = 870 = 0b1101100110 (bits set: 1,2,5,6,8,9).
__device__ __forceinline__ float decrypt_one(uint32_t c, float s) {
    uint32_t b  = c % N2PUB;              // inputs are < N^2; 32-bit magic-div reduce
    uint32_t bm = montmul(b, R2);         // to Montgomery form: b*R
    uint32_t s1 = montmul(bm, bm);        // b^2
    uint32_t rm = s1;                     // bit 1 (first multiply of r==1 is assignment)
    uint32_t s2 = montmul(s1, s1);        // b^4
    rm = montmul(rm, s2);                 // bit 2
    uint32_t s3 = montmul(s2, s2);        // b^8
    uint32_t s4 = montmul(s3, s3);        // b^16
    uint32_t s5 = montmul(s4, s4);        // b^32
    rm = montmul(rm, s5);                 // bit 5
    uint32_t s6 = montmul(s5, s5);        // b^64
    rm = montmul(rm, s6);                 // bit 6
    uint32_t s7 = montmul(s6, s6);        // b^128
    uint32_t s8 = montmul(s7, s7);        // b^256
    rm = montmul(rm, s8);                 // bit 8
    uint32_t s9 = montmul(s8, s8);        // b^512
    rm = montmul(rm, s9);                 // bit 9
    uint32_t r = montmul(rm, 1u);         // out of Montgomery: c^870 mod N^2, in [0, N^2)
    // m = ((r-1)//N * MU) % N ; Python floor-div: r==0 -> (-1)//N = -1 -> (N - MU)
    uint32_t m = (r == 0u) ? (NPUB - MUC)
                           : ((((r - 1u) / NPUB) * MUC) % NPUB);
    return fmaxf((float)m * s, 0.0f);
}

// ---------------- CDNA5 async global->LDS staging ----------------
// Builtin signature (from hipcc diagnostic): param0 = AS(1) int __vector(4)*.
typedef int v4i_t __attribute__((vector_size(16)));
typedef __attribute__((address_space(1))) v4i_t gv4i_t;
typedef __attribute__((address_space(3))) v4i_t lv4i_t;

__device__ __forceinline__ void async_ld_b128(const void* gsrc, void* ldst) {
    __builtin_amdgcn_global_load_async_to_lds_b128(
        (gv4i_t*)(uintptr_t)gsrc, (lv4i_t*)(uintptr_t)ldst, 0, 0);
}

// Each stage: 256 threads x 16B (one longlong2 = 2 ciphertexts per thread) = 512 elems.
// Double-buffered async pipeline, depth 2; ASYNCcnt completes in order per wave.
__global__ __launch_bounds__(256) void paillier_main(
        const long long* __restrict__ c, const float* __restrict__ inv_scale,
        float* __restrict__ out, int nStages) {
    __shared__ longlong2 tiles[2][256];
    const float s = inv_scale[0] / 1000.0f;
    const int tid = threadIdx.x;
    const longlong2* __restrict__ cv = (const longlong2*)c;
    float2* __restrict__ ov = (float2*)out;

    const int s0 = blockIdx.x;
    const int G  = gridDim.x;

    if (s0 < nStages)
        async_ld_b128(&cv[(size_t)s0 * 256 + tid], &tiles[0][tid]);
    if (s0 + G < nStages)
        async_ld_b128(&cv[((size_t)s0 + G) * 256 + tid], &tiles[1][tid]);

    int buf = 0;
    for (int st = s0; st < nStages; st += G, buf ^= 1) {
        // Ensure this stage's tile landed; keep at most the next stage in flight.
        if (st + G < nStages)
            asm volatile("s_wait_asynccnt 0x1" ::: "memory");
        else
            asm volatile("s_wait_asynccnt 0x0" ::: "memory");

        longlong2 v = tiles[buf][tid];                      // ds_load_b128
        asm volatile("s_wait_dscnt 0x0" ::: "memory");      // LDS read done -> slot reusable

        if (st + 2 * G < nStages)                           // prefetch depth-2 into this buffer
            async_ld_b128(&cv[((size_t)st + 2 * (size_t)G) * 256 + tid], &tiles[buf][tid]);

        float2 o;
        o.x = decrypt_one((uint32_t)(uint64_t)v.x, s);
        o.y = decrypt_one((uint32_t)(uint64_t)v.y, s);
        ov[(size_t)st * 256 + tid] = o;                     // global_store_b64
    }
}

// Tail (n not multiple of 512) — plain elementwise path.
__global__ __launch_bounds__(256) void paillier_tail(
        const long long* __restrict__ c, const float* __restrict__ inv_scale,
        float* __restrict__ out, int start, int n) {
    int i = start + blockIdx.x * blockDim.x + threadIdx.x;
    if (i < n) {
        const float s = inv_scale[0] / 1000.0f;
        out[i] = decrypt_one((uint32_t)(uint64_t)c[i], s);
    }
}

extern "C" void kernel_launch(void* const* d_in, const int* in_sizes, int n_in,
                              void* d_out, int out_size, void* d_ws, size_t ws_size,
                              hipStream_t stream) {
    const long long* c   = (const long long*)d_in[0];
    const float*     inv = (const float*)d_in[1];
    float*           out = (float*)d_out;
    const int n = in_sizes[0];            // 8192*4096 = 33,554,432

    const int nStages = n / 512;          // full 512-element stages
    if (nStages > 0) {
        int grid = nStages < 4096 ? nStages : 4096;   // persistent blocks, ~16 stages each
        paillier_main<<<grid, 256, 0, stream>>>(c, inv, out, nStages);
    }
    const int done = nStages * 512;
    if (done < n) {
        int rem = n - done;
        paillier_tail<<<(rem + 255) / 256, 256, 0, stream>>>(c, inv, out, done, n);
    }
}